// MS_Loss_30863634989071
// MI455X (gfx1250) — compile-verified
//
#include <hip/hip_runtime.h>
#include <hip/hip_bf16.h>

// ---- problem constants (from reference) ----
#define BB    8
#define NN    2048      // tokenN
#define N1    2049      // rows (tokenN + 1)
#define DD    302       // feature dim
#define DK    20        // head dim
#define DP    304       // D padded to 19*16 for WMMA N-tiles
#define ROWS  2064      // N1 padded to 129*16 for WMMA M-tiles
#define MT    (ROWS/16) // 129 M-tiles
#define KPAIRS 10       // DK/2 packed k-pairs for Q@M B operand
#define NEG_SLOPE 0.01f

__device__ __forceinline__ float fdx()  { return 2.0f / 2047.0f; }           // grid spacing
#define INV_SQRT_DK 0.22360679774997896f                                      // 1/sqrt(20)

typedef __attribute__((ext_vector_type(2))) float v2f;
typedef __attribute__((ext_vector_type(8))) float v8f;

// ---------------------------------------------------------------------------
// Small MLP [1,32,64,1] with LeakyReLU between layers (shared by 3 param sets)
// ---------------------------------------------------------------------------
__device__ float mlp_eval(float x,
                          const float* __restrict__ w1, const float* __restrict__ b1,
                          const float* __restrict__ w2, const float* __restrict__ b2,
                          const float* __restrict__ w3, const float* __restrict__ b3) {
    float h1[32];
#pragma unroll
    for (int i = 0; i < 32; ++i) {
        float v = w1[i] * x + b1[i];
        h1[i] = v > 0.f ? v : NEG_SLOPE * v;
    }
    float out = b3[0];
    for (int j = 0; j < 64; ++j) {
        float v = b2[j];
#pragma unroll
        for (int i = 0; i < 32; ++i) v += w2[j * 32 + i] * h1[i];
        v = v > 0.f ? v : NEG_SLOPE * v;
        out += w3[j] * v;
    }
    return out;
}

__global__ void k_mlp(const float* kw1, const float* kb1, const float* kw2,
                      const float* kb2, const float* kw3, const float* kb3,
                      const float* fw1, const float* fb1, const float* fw2,
                      const float* fb2, const float* fw3, const float* fb3,
                      const float* gw1, const float* gb1, const float* gw2,
                      const float* gb2, const float* gw3, const float* gb3,
                      const float* gamma, const float* beta,
                      float* weight, float* fnw, float* fnb) {
    int t = blockIdx.x * blockDim.x + threadIdx.x;
    if (t > NN) return;
    if (t < NN) {
        float x = -1.0f + 2.0f * (float)t / (float)(NN - 1);
        weight[t] = mlp_eval(x, kw1, kb1, kw2, kb2, kw3, kb3);
        fnw[t]    = mlp_eval(x, fw1, fb1, fw2, fb2, fw3, fb3);
        fnb[t]    = mlp_eval(x, gw1, gb1, gw2, gb2, gw3, gb3);
    } else {
        fnw[NN] = gamma[0];
        fnb[NN] = beta[0];
    }
}

// ---------------------------------------------------------------------------
// V0[b,t,:] = xy[b,t,:] + P[t,:]   (cols D..DP-1 zeroed for WMMA padding)
// ---------------------------------------------------------------------------
__global__ void k_build_v(const float* __restrict__ xy, const float* __restrict__ P,
                          float* __restrict__ V) {
    size_t idx = (size_t)blockIdx.x * blockDim.x + threadIdx.x;
    if (idx >= (size_t)BB * N1 * DP) return;
    int d = (int)(idx % DP);
    int t = (int)((idx / DP) % N1);
    int b = (int)(idx / ((size_t)DP * N1));
    float v = 0.f;
    if (d < DD) v = xy[(size_t)b * N1 * DD + (size_t)t * DD + d] + P[(size_t)t * DD + d];
    V[(size_t)b * ROWS * DP + (size_t)t * DP + d] = v;
}

// ---------------------------------------------------------------------------
// Pack W[20,302] -> Wp as padded, pair-interleaved W^T:
//   Wp[(k>>1)*64 + col*2 + (k&1)] = (col<20 && k<302) ? W[col,k] : 0
// so the B operand pair (k,k+1) at a given col is one aligned b64 load.
// ---------------------------------------------------------------------------
__global__ void k_pack_w(const float* __restrict__ W, float* __restrict__ Wp) {
    int idx = blockIdx.x * blockDim.x + threadIdx.x;      // over DP*32
    if (idx >= DP * 32) return;
    int col = idx & 31, k = idx >> 5;
    float v = (col < DK && k < DD) ? W[col * DD + k] : 0.f;
    Wp[(k >> 1) * 64 + col * 2 + (k & 1)] = v;
}

// ---------------------------------------------------------------------------
// WMMA projection:  Out[b,t,j] = sum_k V[b,t,k] * W[j,k] + bias[j]
// One wave per 16-row M-tile; both N-tiles (cols 0..31) per wave so each A
// load feeds two v_wmma_f32_16x16x4_f32. All loads unconditional b64.
// ---------------------------------------------------------------------------
__global__ __launch_bounds__(256)
void k_proj_wmma(const float* __restrict__ V, const float* __restrict__ Wp,
                 const float* __restrict__ bias, float* __restrict__ Out) {
    const int wave = blockIdx.x * (blockDim.x >> 5) + (threadIdx.x >> 5);
    const int lane = threadIdx.x & 31;
    if (wave >= BB * MT) return;                // wave-uniform exit
    const int b  = wave / MT;
    const int mt = wave % MT;

    const float* Vb = V + (size_t)b * ROWS * DP;
    const int t0    = mt * 16;
    const int rowA  = t0 + (lane & 15);
    const int khalf = (lane >> 4) * 2;          // ISA A layout: hi lanes hold K+2,K+3
    const int col   = lane & 15;

    const float* Arow = Vb + (size_t)rowA * DP + khalf;
    v8f acc0 = {}, acc1 = {};
    for (int k0 = 0; k0 < DP; k0 += 4) {
        v2f a = *(const v2f*)(Arow + k0);
        const float* bp = Wp + (size_t)((k0 + khalf) >> 1) * 64 + col * 2;
        v2f b0 = *(const v2f*)(bp);
        v2f b1 = *(const v2f*)(bp + 32);
        int kpre = k0 + 64; if (kpre > DP - 4) kpre = DP - 4;
        __builtin_prefetch(Arow + kpre, 0, 1);  // global_prefetch_b8 on A stream
        acc0 = __builtin_amdgcn_wmma_f32_16x16x4_f32(false, a, false, b0,
                                                     (short)0, acc0, false, false);
        acc1 = __builtin_amdgcn_wmma_f32_16x16x4_f32(false, a, false, b1,
                                                     (short)0, acc1, false, false);
    }
    const float bc0 = bias[col];
    const float bc1 = (col + 16 < DK) ? bias[col + 16] : 0.f;
    const int rbase = t0 + ((lane >> 4) << 3);  // D layout: VGPR r -> M=r (+8 hi lanes)
    float* Ob = Out + (size_t)b * ROWS * DK;
#pragma unroll
    for (int rr = 0; rr < 8; ++rr) {
        int row = rbase + rr;
        if (row < N1) {
            Ob[(size_t)row * DK + col] = acc0[rr] + bc0;
            if (col + 16 < DK)
                Ob[(size_t)row * DK + col + 16] = acc1[rr] + bc1;
        }
    }
}

// ---------------------------------------------------------------------------
// Mmat[b,j,d] = sum_{t<N1} K[b,t,j] * V[b,t,d], stored pair-interleaved:
//   Mp[b][j>>1][d][j&1]   (so Q@M's B pair (j,j+1) is one aligned b64 load)
// ---------------------------------------------------------------------------
__global__ void k_ktv(const float* __restrict__ K, const float* __restrict__ V,
                      float* __restrict__ Mp) {
    int idx = blockIdx.x * blockDim.x + threadIdx.x;
    if (idx >= BB * DK * DP) return;
    int d = idx % DP;
    int j = (idx / DP) % DK;
    int b = idx / (DP * DK);
    const float* Kb = K + (size_t)b * ROWS * DK;
    const float* Vb = V + (size_t)b * ROWS * DP;
    float acc = 0.f;
    for (int t = 0; t < N1; ++t)
        acc = fmaf(Kb[(size_t)t * DK + j], Vb[(size_t)t * DP + d], acc);
    Mp[(size_t)b * KPAIRS * DP * 2 + (size_t)(j >> 1) * DP * 2 + d * 2 + (j & 1)] = acc;
}

// ---------------------------------------------------------------------------
// WMMA recombine: mid[b,t,d] = scale * sum_j Q[b,t,j] * Mmat[b,j,d]
// 5 K-steps (dk=20); unconditional b64 loads for A and packed-B.
// ---------------------------------------------------------------------------
__global__ __launch_bounds__(256)
void k_qm_wmma(const float* __restrict__ Q, const float* __restrict__ Mp,
               float* __restrict__ mid) {
    const int wave = blockIdx.x * (blockDim.x >> 5) + (threadIdx.x >> 5);
    const int lane = threadIdx.x & 31;
    const int NT = DP / 16;                     // 19
    if (wave >= BB * MT * NT) return;
    int b  = wave / (MT * NT);
    int r  = wave % (MT * NT);
    int mt = r / NT, nt = r % NT;

    const float* Qb = Q + (size_t)b * ROWS * DK;
    const float* Mb = Mp + (size_t)b * KPAIRS * DP * 2;
    const int t0 = mt * 16, n0 = nt * 16;
    const int rowA  = t0 + (lane & 15);
    const int khalf = (lane >> 4) * 2;
    const int colB  = n0 + (lane & 15);

    const float* Arow = Qb + (size_t)rowA * DK + khalf;
    v8f acc = {};
#pragma unroll
    for (int k0 = 0; k0 < DK; k0 += 4) {
        v2f a  = *(const v2f*)(Arow + k0);
        v2f bm = *(const v2f*)(Mb + (size_t)((k0 + khalf) >> 1) * DP * 2 + colB * 2);
        acc = __builtin_amdgcn_wmma_f32_16x16x4_f32(false, a, false, bm,
                                                    (short)0, acc, false, false);
    }
    const float scale = fdx() * INV_SQRT_DK;
    int rbase = t0 + ((lane >> 4) << 3);
#pragma unroll
    for (int rr = 0; rr < 8; ++rr) {
        int row = rbase + rr;
        if (row < N1)
            mid[(size_t)b * ROWS * DP + (size_t)row * DP + colB] = scale * acc[rr];
    }
}

// mean over rows: mean[b,d] = (1/N1) sum_t mid[b,t,d]
__global__ void k_mean(const float* __restrict__ mid, float* __restrict__ mean) {
    int idx = blockIdx.x * blockDim.x + threadIdx.x;
    if (idx >= BB * DP) return;
    int d = idx % DP, b = idx / DP;
    float s = 0.f;
    const float* mb = mid + (size_t)b * ROWS * DP;
    for (int t = 0; t < N1; ++t) s += mb[(size_t)t * DP + d];
    mean[idx] = s * (1.0f / (float)N1);
}

// in-place: V = fnb[t] + (mid - mean)*fnw[t] + V   (cols >= D forced to 0)
__global__ void k_update(float* __restrict__ V, const float* __restrict__ mid,
                         const float* __restrict__ mean, const float* __restrict__ fnw,
                         const float* __restrict__ fnb) {
    size_t idx = (size_t)blockIdx.x * blockDim.x + threadIdx.x;
    if (idx >= (size_t)BB * N1 * DP) return;
    int d = (int)(idx % DP);
    int t = (int)((idx / DP) % N1);
    int b = (int)(idx / ((size_t)DP * N1));
    size_t vi = (size_t)b * ROWS * DP + (size_t)t * DP + d;
    if (d < DD)
        V[vi] = fnb[t] + (mid[vi] - mean[b * DP + d]) * fnw[t] + V[vi];
    else
        V[vi] = 0.f;
}

// g[b,j] = invsqrtdk * ( dx^2 * sum_{t<N} weight[t]*Q1[b,t,j] + dx*kernelf*Q1[b,N,j] )
__global__ void k_gvec(const float* __restrict__ Q, const float* __restrict__ weight,
                       const float* __restrict__ kernelf, float* __restrict__ g) {
    int b = blockIdx.x, j = threadIdx.x;
    if (j >= DK) return;
    const float* Qb = Q + (size_t)b * ROWS * DK;
    float s = 0.f;
    for (int t = 0; t < NN; ++t) s = fmaf(weight[t], Qb[(size_t)t * DK + j], s);
    float dx = fdx();
    g[b * DK + j] = INV_SQRT_DK * (dx * dx * s + dx * kernelf[0] * Qb[(size_t)NN * DK + j]);
}

// coef[b,m] = sum_j g[b,j] * K1[b,m,j]
__global__ void k_coef(const float* __restrict__ K, const float* __restrict__ g,
                       float* __restrict__ coef) {
    int idx = blockIdx.x * blockDim.x + threadIdx.x;
    if (idx >= BB * NN) return;
    int m = idx % NN, b = idx / NN;
    const float* Kb = K + (size_t)b * ROWS * DK + (size_t)m * DK;
    float s = 0.f;
#pragma unroll
    for (int j = 0; j < DK; ++j) s = fmaf(g[b * DK + j], Kb[j], s);
    coef[idx] = s;
}

// out[b,d] = sum_{m<N} coef[b,m] * xy[b,m,d]
__global__ void k_out(const float* __restrict__ xy, const float* __restrict__ coef,
                      float* __restrict__ out) {
    int idx = blockIdx.x * blockDim.x + threadIdx.x;
    if (idx >= BB * DD) return;
    int d = idx % DD, b = idx / DD;
    const float* xb = xy + (size_t)b * N1 * DD;
    const float* cb = coef + (size_t)b * NN;
    float s = 0.f;
    for (int m = 0; m < NN; ++m) s = fmaf(cb[m], xb[(size_t)m * DD + d], s);
    out[idx] = s;
}

// ---------------------------------------------------------------------------
extern "C" void kernel_launch(void* const* d_in, const int* in_sizes, int n_in,
                              void* d_out, int out_size, void* d_ws, size_t ws_size,
                              hipStream_t stream) {
    auto F = [&](int i) { return (const float*)d_in[i]; };

    const float *xy, *P;
    const float *kp[6], *fwp[6], *fbp[6];
    const float *kernelf, *gamma, *beta;
    const float *qw0, *qb0, *qw1, *qb1, *kw0, *kb0, *kw1, *kb1;

    if (n_in >= 32) {
        // tree-flattened leaves, dict order
        int i = 0;
        xy = F(i++); P = F(i++); i++; /* delta */
        for (int k = 0; k < 6; ++k) kp[k]  = F(i++);
        for (int k = 0; k < 6; ++k) fwp[k] = F(i++);
        for (int k = 0; k < 6; ++k) fbp[k] = F(i++);
        kernelf = F(i++); gamma = F(i++); beta = F(i++);
        qw0 = F(i++); qb0 = F(i++); qw1 = F(i++); qb1 = F(i++);
        kw0 = F(i++); kb0 = F(i++); kw1 = F(i++); kb1 = F(i++);
    } else {
        // one concatenated buffer per dict key
        xy = F(0); P = F(1);
        const float* kb = F(3);
        kp[0] = kb; kp[1] = kb + 32; kp[2] = kb + 64; kp[3] = kb + 2112; kp[4] = kb + 2176; kp[5] = kb + 2240;
        const float* fw = F(4);
        fwp[0] = fw; fwp[1] = fw + 32; fwp[2] = fw + 64; fwp[3] = fw + 2112; fwp[4] = fw + 2176; fwp[5] = fw + 2240;
        const float* fb = F(5);
        fbp[0] = fb; fbp[1] = fb + 32; fbp[2] = fb + 64; fbp[3] = fb + 2112; fbp[4] = fb + 2176; fbp[5] = fb + 2240;
        kernelf = F(6); gamma = F(7); beta = F(8);
        const float* q = F(9);
        qw0 = q; qb0 = q + 6040; qw1 = q + 6060; qb1 = q + 12100;
        const float* kk = F(10);
        kw0 = kk; kb0 = kk + 6040; kw1 = kk + 6060; kb1 = kk + 12100;
    }

    // workspace layout (floats)
    float* ws = (float*)d_ws;
    size_t o = 0;
    float* w_weight = ws + o; o += 2048;
    float* w_fnw    = ws + o; o += 2064;
    float* w_fnb    = ws + o; o += 2064;
    float* w_g      = ws + o; o += 256;
    float* w_coef   = ws + o; o += (size_t)BB * NN;
    float* w_mean   = ws + o; o += (size_t)BB * DP;
    float* w_Wp     = ws + o; o += (size_t)DP * 32;           // packed W^T
    float* w_M      = ws + o; o += (size_t)BB * KPAIRS * DP * 2;
    float* w_Q      = ws + o; o += (size_t)BB * ROWS * DK;
    float* w_K      = ws + o; o += (size_t)BB * ROWS * DK;
    float* w_V      = ws + o; o += (size_t)BB * ROWS * DP;
    float* w_mid    = ws + o; o += (size_t)BB * ROWS * DP;
    (void)ws_size; (void)in_sizes; (void)out_size;

    const int elemsV    = BB * N1 * DP;
    const int projWaves = BB * MT;
    const int qmWaves   = BB * MT * (DP / 16);
    const int packGrid  = (DP * 32 + 255) / 256;

    // 1. grid MLPs -> weight / fn_weight / fn_bias
    k_mlp<<<(N1 + 255) / 256, 256, 0, stream>>>(
        kp[0], kp[1], kp[2], kp[3], kp[4], kp[5],
        fwp[0], fwp[1], fwp[2], fwp[3], fwp[4], fwp[5],
        fbp[0], fbp[1], fbp[2], fbp[3], fbp[4], fbp[5],
        gamma, beta, w_weight, w_fnw, w_fnb);

    // 2. V0 = xy + P (padded)
    k_build_v<<<(elemsV + 255) / 256, 256, 0, stream>>>(xy, P, w_V);

    // 3. layer-0 projections (pack weights, then WMMA f32 16x16x4)
    k_pack_w<<<packGrid, 256, 0, stream>>>(qw0, w_Wp);
    k_proj_wmma<<<(projWaves + 7) / 8, 256, 0, stream>>>(w_V, w_Wp, qb0, w_Q);
    k_pack_w<<<packGrid, 256, 0, stream>>>(kw0, w_Wp);
    k_proj_wmma<<<(projWaves + 7) / 8, 256, 0, stream>>>(w_V, w_Wp, kb0, w_K);

    // 4. Mmat = K^T V   (associativity trick: no NxN attention matrix)
    k_ktv<<<(BB * DK * DP + 255) / 256, 256, 0, stream>>>(w_K, w_V, w_M);

    // 5. mid = (dx/sqrt(dk)) * Q @ Mmat  (WMMA)
    k_qm_wmma<<<(qmWaves + 7) / 8, 256, 0, stream>>>(w_Q, w_M, w_mid);

    // 6. row mean; 7. residual/affine update (in place on V)
    k_mean<<<(BB * DP + 255) / 256, 256, 0, stream>>>(w_mid, w_mean);
    k_update<<<(elemsV + 255) / 256, 256, 0, stream>>>(w_V, w_mid, w_mean, w_fnw, w_fnb);

    // 8. final-layer projections (WMMA)
    k_pack_w<<<packGrid, 256, 0, stream>>>(qw1, w_Wp);
    k_proj_wmma<<<(projWaves + 7) / 8, 256, 0, stream>>>(w_V, w_Wp, qb1, w_Q);
    k_pack_w<<<packGrid, 256, 0, stream>>>(kw1, w_Wp);
    k_proj_wmma<<<(projWaves + 7) / 8, 256, 0, stream>>>(w_V, w_Wp, kb1, w_K);

    // 9-11. collapsed output reductions
    k_gvec<<<BB, 32, 0, stream>>>(w_Q, w_weight, kernelf, w_g);
    k_coef<<<(BB * NN + 255) / 256, 256, 0, stream>>>(w_K, w_g, w_coef);
    k_out<<<(BB * DD + 255) / 256, 256, 0, stream>>>(xy, w_coef, (float*)d_out);
}